// FitTorch_49228915147444
// MI455X (gfx1250) — compile-verified
//
#include <hip/hip_runtime.h>
#include <hip/hip_bf16.h>

#define ND 64
#define HH 128

typedef float v2f __attribute__((ext_vector_type(2)));
typedef float v8f __attribute__((ext_vector_type(8)));

__device__ __forceinline__ v8f wmma4(v2f a, v2f b, v8f c) {
    // D = A(16x4,f32) * B(4x16,f32) + C(16x16,f32)
    return __builtin_amdgcn_wmma_f32_16x16x4_f32(
        /*neg_a=*/false, a, /*neg_b=*/false, b,
        /*c_mod=*/(short)0, c, /*reuse_a=*/false, /*reuse_b=*/false);
}

// sigmoid via v_exp_f32 + v_rcp_f32 (avoids the IEEE div_scale/div_fixup chain)
__device__ __forceinline__ float fast_sigmoid(float v) {
    return __builtin_amdgcn_rcpf(1.0f + __expf(-v));
}

__global__ __launch_bounds__(256) void zero_kernel(float* __restrict__ p, int n) {
    int i = blockIdx.x * blockDim.x + threadIdx.x;
    if (i < n) p[i] = 0.0f;
}

// One block = 16 atoms, 4 waves. Fused MLP forward + energy + backward dEdD.
__global__ __launch_bounds__(128) void mlp_kernel(
    const float* __restrict__ x, const int* __restrict__ indices,
    const float* __restrict__ W1, const float* __restrict__ b1,
    const float* __restrict__ W2, const float* __restrict__ b2,
    const float* __restrict__ W3, const float* __restrict__ b3,
    float* __restrict__ energy_out, float* __restrict__ dEdD)
{
    __shared__ float buf1[16 * HH];   // h1, later overwritten with d1
    __shared__ float buf2[16 * HH];   // h2, later overwritten with d2
    __shared__ float e_acc[16];

    const int tid   = threadIdx.x;
    const int wave  = tid >> 5;           // 0..3
    const int lane  = tid & 31;
    const int lr    = lane & 15;          // row (A) / col (B,C,D) within tile
    const int kh    = (lane >> 4) * 2;    // K-pair select for A/B operands
    const int mh    = (lane >> 4) << 3;   // row offset for C/D operands
    const int atom0 = blockIdx.x * 16;

    if (tid < 16) e_acc[tid] = 0.0f;

    const int n0 = 2 * wave, n1 = 2 * wave + 1;   // this wave's column tiles (of 8)

    // ---------------- layer 1: h1 = sigmoid(x @ W1 + b1) ----------------
    {
        v8f acc0, acc1;
        const float bb0 = b1[n0 * 16 + lr], bb1 = b1[n1 * 16 + lr];
        for (int i = 0; i < 8; ++i) { acc0[i] = bb0; acc1[i] = bb1; }
        for (int kt = 0; kt < ND / 4; ++kt) {
            const int k = kt * 4 + kh;
            v2f a;  a[0]  = x[(atom0 + lr) * ND + k];
                    a[1]  = x[(atom0 + lr) * ND + k + 1];
            v2f bm0; bm0[0] = W1[k * HH + n0 * 16 + lr];
                     bm0[1] = W1[(k + 1) * HH + n0 * 16 + lr];
            v2f bm1; bm1[0] = W1[k * HH + n1 * 16 + lr];
                     bm1[1] = W1[(k + 1) * HH + n1 * 16 + lr];
            acc0 = wmma4(a, bm0, acc0);
            acc1 = wmma4(a, bm1, acc1);
        }
        for (int i = 0; i < 8; ++i) {
            const int m = i + mh;
            buf1[m * HH + n0 * 16 + lr] = fast_sigmoid(acc0[i]);
            buf1[m * HH + n1 * 16 + lr] = fast_sigmoid(acc1[i]);
        }
    }
    __syncthreads();

    // ---------------- layer 2: h2 = sigmoid(h1 @ W2 + b2) ----------------
    {
        v8f acc0, acc1;
        const float bb0 = b2[n0 * 16 + lr], bb1 = b2[n1 * 16 + lr];
        for (int i = 0; i < 8; ++i) { acc0[i] = bb0; acc1[i] = bb1; }
        for (int kt = 0; kt < HH / 4; ++kt) {
            const int k = kt * 4 + kh;
            v2f a;  a[0]  = buf1[lr * HH + k];
                    a[1]  = buf1[lr * HH + k + 1];
            v2f bm0; bm0[0] = W2[k * HH + n0 * 16 + lr];
                     bm0[1] = W2[(k + 1) * HH + n0 * 16 + lr];
            v2f bm1; bm1[0] = W2[k * HH + n1 * 16 + lr];
                     bm1[1] = W2[(k + 1) * HH + n1 * 16 + lr];
            acc0 = wmma4(a, bm0, acc0);
            acc1 = wmma4(a, bm1, acc1);
        }
        for (int i = 0; i < 8; ++i) {
            const int m = i + mh;
            buf2[m * HH + n0 * 16 + lr] = fast_sigmoid(acc0[i]);
            buf2[m * HH + n1 * 16 + lr] = fast_sigmoid(acc1[i]);
        }
    }
    __syncthreads();

    // ---------------- energy: e[m] = h2[m,:] . W3 + b3, scatter by conf ----------------
    {
        const int m = tid & 15, c = tid >> 4;  // 8 chunks of 16 per atom
        float p = 0.0f;
        for (int k = c * 16; k < c * 16 + 16; ++k) p = fmaf(buf2[m * HH + k], W3[k], p);
        atomicAdd(&e_acc[m], p);
    }
    __syncthreads();
    if (tid < 16) {
        const float e = e_acc[tid] + b3[0];
        atomicAdd(&energy_out[indices[atom0 + tid]], e);
    }
    __syncthreads();

    // ---------------- d2 = W3 * h2 * (1 - h2), in place in buf2 ----------------
    for (int idx = tid; idx < 16 * HH; idx += 128) {
        const int k = idx & (HH - 1);
        const float h = buf2[idx];
        buf2[idx] = W3[k] * h * (1.0f - h);
    }
    __syncthreads();

    // ---------------- g1 = d2 @ W2^T ; d1 = g1 * h1 * (1 - h1), into buf1 ----------------
    {
        v8f acc0 = {}, acc1 = {};
        for (int kt = 0; kt < HH / 4; ++kt) {
            const int k = kt * 4 + kh;
            v2f a;  a[0]  = buf2[lr * HH + k];
                    a[1]  = buf2[lr * HH + k + 1];
            v2f bm0; bm0[0] = W2[(n0 * 16 + lr) * HH + k];       // W2^T
                     bm0[1] = W2[(n0 * 16 + lr) * HH + k + 1];
            v2f bm1; bm1[0] = W2[(n1 * 16 + lr) * HH + k];
                     bm1[1] = W2[(n1 * 16 + lr) * HH + k + 1];
            acc0 = wmma4(a, bm0, acc0);
            acc1 = wmma4(a, bm1, acc1);
        }
        for (int i = 0; i < 8; ++i) {
            const int m = i + mh;
            float h = buf1[m * HH + n0 * 16 + lr];
            buf1[m * HH + n0 * 16 + lr] = acc0[i] * h * (1.0f - h);
            h = buf1[m * HH + n1 * 16 + lr];
            buf1[m * HH + n1 * 16 + lr] = acc1[i] * h * (1.0f - h);
        }
    }
    __syncthreads();

    // ---------------- dEdD = d1 @ W1^T (16 x 64), one col-tile per wave ----------------
    {
        const int n = wave;  // 0..3
        v8f acc = {};
        for (int kt = 0; kt < HH / 4; ++kt) {
            const int k = kt * 4 + kh;
            v2f a;  a[0] = buf1[lr * HH + k];
                    a[1] = buf1[lr * HH + k + 1];
            v2f bm; bm[0] = W1[(n * 16 + lr) * HH + k];          // W1^T
                    bm[1] = W1[(n * 16 + lr) * HH + k + 1];
            acc = wmma4(a, bm, acc);
        }
        for (int i = 0; i < 8; ++i) {
            const int m = i + mh;
            dEdD[(size_t)(atom0 + m) * ND + n * 16 + lr] = acc[i];
        }
    }
}

// One wave per pair (3 derivative rows share one gathered dEdD row).
__global__ __launch_bounds__(256) void forces_kernel(
    const float* __restrict__ xd, const int* __restrict__ unique_i,
    const int* __restrict__ unique_j, const int* __restrict__ xd_indx,
    const float* __restrict__ dEdD, float* __restrict__ forces, int npairs)
{
    const int lane = threadIdx.x & 31;
    const int gw = (blockIdx.x * blockDim.x + threadIdx.x) >> 5;
    const int nw = (gridDim.x * blockDim.x) >> 5;

    for (int p = gw; p < npairs; p += nw) {
        const int r0 = 3 * p;
        const int ia = unique_i[r0];
        const int ja = unique_j[r0];
        const float2 dv = ((const float2*)(dEdD + (size_t)ia * ND))[lane];

        if (p + nw < npairs)  // pull next pair's xd block toward the caches
            __builtin_prefetch(xd + (size_t)3 * (p + nw) * ND, 0, 1);

        for (int a = 0; a < 3; ++a) {
            const float2 xv = ((const float2*)(xd + (size_t)(r0 + a) * ND))[lane];
            float s = fmaf(xv.x, dv.x, xv.y * dv.y);
            for (int off = 16; off > 0; off >>= 1) s += __shfl_down(s, off, 32);
            if (lane == 0) {
                const int axis = xd_indx[(size_t)(r0 + a) * 3 + 2];
                atomicAdd(&forces[ja * 3 + axis], s);
            }
        }
    }
}

extern "C" void kernel_launch(void* const* d_in, const int* in_sizes, int n_in,
                              void* d_out, int out_size, void* d_ws, size_t ws_size,
                              hipStream_t stream) {
    const float* x        = (const float*)d_in[0];
    const float* xd       = (const float*)d_in[1];
    const int*   indices  = (const int*)d_in[2];
    const int*   xd_indx  = (const int*)d_in[5];
    const int*   unique_i = (const int*)d_in[6];
    const int*   unique_j = (const int*)d_in[7];
    const float* W1       = (const float*)d_in[8];
    const float* b1       = (const float*)d_in[9];
    const float* W2       = (const float*)d_in[10];
    const float* b2       = (const float*)d_in[11];
    const float* W3       = (const float*)d_in[12];
    const float* b3       = (const float*)d_in[13];

    const int natoms = in_sizes[0] / ND;          // 20000
    const int nconf  = in_sizes[3];               // 500
    const int npairs = in_sizes[6] / 3;           // 400000

    float* out   = (float*)d_out;                 // [0,nconf) energies, then forces
    float* dEdD  = (float*)d_ws;                  // natoms*ND floats scratch

    zero_kernel<<<(out_size + 255) / 256, 256, 0, stream>>>(out, out_size);
    mlp_kernel<<<natoms / 16, 128, 0, stream>>>(x, indices, W1, b1, W2, b2, W3, b3,
                                                out, dEdD);
    forces_kernel<<<2048, 256, 0, stream>>>(xd, unique_i, unique_j, xd_indx,
                                            dEdD, out + nconf, npairs);
}